// PyramidAdaptiveBlockSparseAttnTrain_83923660963896
// MI455X (gfx1250) — compile-verified
//
#include <hip/hip_runtime.h>
#include <hip/hip_bf16.h>
#include <math.h>

#define NH   8
#define SEQ  2048
#define DIM  128
#define BLK  128     // BS
#define SBK  16      // SEQ/BLK
#define NCH  128     // SEQ/STRIDE

typedef float v2f __attribute__((ext_vector_type(2)));
typedef float v8f __attribute__((ext_vector_type(8)));

__device__ __forceinline__ v8f wmma4(v2f a, v2f b, v8f c) {
  // V_WMMA_F32_16X16X4_F32 : D = A(16x4) x B(4x16) + C(16x16), fp32
  return __builtin_amdgcn_wmma_f32_16x16x4_f32(false, a, false, b, (short)0, c,
                                               false, false);
}

// ---------------- Stage 1a: stride-16 pooling of q,k -> qg,kg [H,128,128] ---
__global__ void pool16_kernel(const float* __restrict__ q,
                              const float* __restrict__ k,
                              float* __restrict__ qg, float* __restrict__ kg) {
  int bid = blockIdx.x;            // h*NCH + c
  int h = bid >> 7, c = bid & 127;
  int d = threadIdx.x;             // 0..127
  const float* qp = q + (size_t)(h * SEQ + c * 16) * DIM + d;
  const float* kp = k + (size_t)(h * SEQ + c * 16) * DIM + d;
  float sq = 0.f, sk = 0.f;
#pragma unroll
  for (int s = 0; s < 16; s++) { sq += qp[s * DIM]; sk += kp[s * DIM]; }
  qg[(h * NCH + c) * DIM + d] = sq;
  kg[(h * NCH + c) * DIM + d] = sk;
}

// ---------------- Stage 1b: causal softmax of qg@kg^T rows -> probs ---------
__global__ void chunk_softmax_kernel(const float* __restrict__ qg,
                                     const float* __restrict__ kg,
                                     float* __restrict__ probs) {
  __shared__ float qrow[DIM];
  __shared__ float red[128];
  int bid = blockIdx.x;            // h*NCH + c
  int h = bid >> 7, c = bid & 127;
  int j = threadIdx.x;             // 0..127 key chunk
  qrow[j] = qg[(h * NCH + c) * DIM + j];
  __syncthreads();
  const float* kr = kg + (size_t)(h * NCH + j) * DIM;
  float dot = 0.f;
  for (int d = 0; d < DIM; d++) dot += qrow[d] * kr[d];
  float x = (j <= c) ? dot * 0.08838834764831845f : -INFINITY;  // 1/sqrt(128)
  red[j] = x;  __syncthreads();
  for (int s = 64; s > 0; s >>= 1) { if (j < s) red[j] = fmaxf(red[j], red[j + s]); __syncthreads(); }
  float mx = red[0];  __syncthreads();
  float p = (j <= c) ? __expf(x - mx) : 0.0f;
  red[j] = p;  __syncthreads();
  for (int s = 64; s > 0; s >>= 1) { if (j < s) red[j] += red[j + s]; __syncthreads(); }
  probs[(h * NCH + c) * NCH + j] = p / red[0];
}

// ---------------- Stage 1c: pool probs to block importance est [H,16,16] ----
__global__ void est_pool_kernel(const float* __restrict__ probs,
                                float* __restrict__ est) {
  int h = blockIdx.x;              // 0..7
  int idx = threadIdx.x;           // 0..255
  int ib = idx >> 4, jb = idx & 15;
  float acc = 0.f;
#pragma unroll
  for (int ri = 0; ri < 8; ri++)
#pragma unroll
    for (int rj = 0; rj < 8; rj++)
      acc += probs[(h * NCH + ib * 8 + ri) * NCH + jb * 8 + rj];
  est[(h * SBK + ib) * SBK + jb] = acc * 0.125f;   // sum over rj, mean over ri
}

// ---------------- Stage 3: hierarchical cosine similarity -> sim [H,16] -----
__global__ void ksim_kernel(const float* __restrict__ k, int* __restrict__ sim) {
  __shared__ float red[64];
  int bid = blockIdx.x;            // h*SBK + jb
  int h = bid >> 4, jb = bid & 15;
  int t = threadIdx.x;             // 0..63
  const float* kb = k + (size_t)(h * SEQ + jb * BLK) * DIM;

  float dot = 0.f, na = 0.f, nb = 0.f;
  {
    const float* a = kb + (size_t)(2 * t) * DIM;
    for (int d = 0; d < DIM; d++) {
      float av = a[d], bv = a[DIM + d];
      dot += av * bv; na += av * av; nb += bv * bv;
    }
  }
  red[t] = dot / (sqrtf(na) * sqrtf(nb) + 1e-6f);
  __syncthreads();
  for (int s = 32; s > 0; s >>= 1) { if (t < s) red[t] += red[t + s]; __syncthreads(); }
  float s2 = red[0] * (1.0f / 64.0f);
  __syncthreads();

  float c4 = 0.f;
  if (t < 32) {
    dot = 0.f; na = 0.f; nb = 0.f;
    const float* r0 = kb + (size_t)(4 * t) * DIM;
    for (int d = 0; d < DIM; d++) {
      float av = 0.5f * (r0[d] + r0[DIM + d]);
      float bv = 0.5f * (r0[2 * DIM + d] + r0[3 * DIM + d]);
      dot += av * bv; na += av * av; nb += bv * bv;
    }
    c4 = dot / (sqrtf(na) * sqrtf(nb) + 1e-6f);
  }
  red[t] = (t < 32) ? c4 : 0.0f;  __syncthreads();
  for (int s = 32; s > 0; s >>= 1) { if (t < s) red[t] += red[t + s]; __syncthreads(); }
  float s4 = red[0] * (1.0f / 32.0f);
  __syncthreads();

  float c8 = 0.f;
  if (t < 16) {
    dot = 0.f; na = 0.f; nb = 0.f;
    const float* r0 = kb + (size_t)(8 * t) * DIM;
    for (int d = 0; d < DIM; d++) {
      float av = 0.25f * (r0[d] + r0[DIM + d] + r0[2 * DIM + d] + r0[3 * DIM + d]);
      float bv = 0.25f * (r0[4 * DIM + d] + r0[5 * DIM + d] + r0[6 * DIM + d] + r0[7 * DIM + d]);
      dot += av * bv; na += av * av; nb += bv * bv;
    }
    c8 = dot / (sqrtf(na) * sqrtf(nb) + 1e-6f);
  }
  red[t] = (t < 16) ? c8 : 0.0f;  __syncthreads();
  for (int s = 32; s > 0; s >>= 1) { if (t < s) red[t] += red[t + s]; __syncthreads(); }
  float s8 = red[0] * (1.0f / 16.0f);

  if (t == 0)
    sim[h * SBK + jb] = (s2 > 0.75f) ? ((s4 > 0.7f) ? ((s8 > 0.7f) ? 8 : 4) : 2) : 1;
}

// ---------------- Stage 2: rank thresholds -> mask [H,16,16] ----------------
__global__ void mask_kernel(const float* __restrict__ est,
                            const int* __restrict__ sim, int* __restrict__ msk) {
  int t = threadIdx.x;
  if (t >= NH * SBK) return;
  int h = t >> 4, iq = t & 15;
  float e[16];
#pragma unroll
  for (int j = 0; j < 16; j++) e[j] = est[(h * SBK + iq) * SBK + j];
  float L = (float)(iq + 1);
  int t0 = (int)(L * 0.05f), t1 = (int)(L * 0.15f);
  int t2 = (int)(L * 0.25f), t3 = (int)(L * 0.50f);
#pragma unroll
  for (int j = 0; j < 16; j++) {
    int rank = 0;   // stable argsort of -e: greater first, ties by index
    for (int j2 = 0; j2 < 16; j2++)
      rank += (e[j2] > e[j]) || ((e[j2] == e[j]) && (j2 < j));
    int mv = (rank < t0) ? 1 : (rank < t1) ? 2 : (rank < t2) ? 4 : (rank < t3) ? 8 : 0;
    if (j >= SBK - 4) mv = 1;       // last ceil(512/128)=4 columns
    if (iq >= SBK - 4) mv = 1;      // last 4 rows
    if (j > iq) mv = 0;             // strict upper triangle
    if (j == iq) mv = 1;            // diagonal
    if (j == 0) mv = 1;             // first column
    int sv = sim[h * SBK + j];
    msk[(h * SBK + iq) * SBK + j] = (mv < sv) ? mv : sv;
  }
}

// ---------------- Stage 4: fused multi-res block-sparse flash attention -----
__global__ void __launch_bounds__(256)
attn_kernel(const float* __restrict__ q, const float* __restrict__ k,
            const float* __restrict__ v, const int* __restrict__ msk,
            float* __restrict__ out) {
  __shared__ float Klds[16 * DIM];      // pooled K tile (16 pooled keys x 128)
  __shared__ float Vlds[16 * DIM];      // pooled V tile
  __shared__ float Plds[8][16 * 16];    // per-wave P staging (C->A relayout)

  const int bid = blockIdx.x;           // h*SBK + ib
  const int h = bid >> 4, ib = bid & 15;
  const int w = threadIdx.x >> 5;       // wave id: 16-query tile
  const int lane = threadIdx.x & 31;
  const int hi = lane >> 4;             // lane half
  const int lo = lane & 15;

  // Q A-fragments (ISA 16x4 layout: M = lo, K = 4s + v + 2*hi)
  const float* qrow = q + (size_t)(h * SEQ + ib * BLK + w * 16 + lo) * DIM;
  v2f aq[32];
#pragma unroll
  for (int s = 0; s < 32; s++) {
    aq[s].x = qrow[4 * s + 2 * hi];
    aq[s].y = qrow[4 * s + 2 * hi + 1];
  }

  v8f O[8];                             // 16x128 output tile, C layout
#pragma unroll
  for (int g = 0; g < 8; g++) O[g] = {};
  float mrow[8], srow[8];               // per-row running max / sum (row = r+8*hi)
#pragma unroll
  for (int r = 0; r < 8; r++) { mrow[r] = -INFINITY; srow[r] = 0.0f; }

  const float scale = 0.08838834764831845f;  // 1/sqrt(128)

  for (int j = 0; j <= ib; j++) {
    const int lv = msk[(h * SBK + ib) * SBK + j];   // uniform across block
    if (lv == 0) continue;
    const int ntiles = (BLK / lv) >> 4;             // 8/4/2/1 for lv 1/2/4/8
    const float loglv = (lv == 1) ? 0.0f : (lv == 2) ? 0.6931471805599453f
                      : (lv == 4) ? 1.3862943611198906f : 2.0794415416798357f;
    const float invlv = 1.0f / (float)lv;

    for (int tt = 0; tt < ntiles; tt++) {
      __syncthreads();                  // previous tile fully consumed
      // cooperative on-the-fly pooled K/V tile (mean over lv rows)
      for (int idx = threadIdx.x; idx < 16 * DIM; idx += 256) {
        int trow = idx >> 7, d = idx & 127;
        size_t srcrow = (size_t)(h * SEQ + j * BLK + (tt * 16 + trow) * lv);
        const float* kp = k + srcrow * DIM + d;
        const float* vp = v + srcrow * DIM + d;
        float sk = 0.f, sv = 0.f;
        for (int u = 0; u < lv; u++) { sk += kp[u * DIM]; sv += vp[u * DIM]; }
        Klds[idx] = sk * invlv;
        Vlds[idx] = sv * invlv;
      }
      __syncthreads();

      // S = Q * Kpool^T : 32 chained fp32 WMMAs over K=128
      v8f Sf = {};
#pragma unroll
      for (int s = 0; s < 32; s++) {
        v2f bk;                         // B 4x16: K = 4s+v+2*hi, N = lo
        bk.x = Klds[lo * DIM + 4 * s + 2 * hi];
        bk.y = Klds[lo * DIM + 4 * s + 2 * hi + 1];
        Sf = wmma4(aq[s], bk, Sf);
      }

      // mask + online softmax (C layout: row = r+8*hi, col = lo)
      float x[8], tm[8];
#pragma unroll
      for (int r = 0; r < 8; r++) {
        int qloc = w * 16 + r + 8 * hi;             // query index in block
        int pos = (tt * 16 + lo) * lv + lv - 1;     // rightmost pooled token
        bool valid = (j < ib) || (pos <= qloc);
        x[r] = valid ? (Sf[r] * scale + loglv) : -INFINITY;
        tm[r] = x[r];
      }
#pragma unroll
      for (int off = 1; off <= 8; off <<= 1)
#pragma unroll
        for (int r = 0; r < 8; r++)
          tm[r] = fmaxf(tm[r], __shfl_xor(tm[r], off, 32));

      float p[8], alpha[8];
#pragma unroll
      for (int r = 0; r < 8; r++) {
        float mnew = fmaxf(mrow[r], tm[r]);
        alpha[r] = (mnew == -INFINITY) ? 1.0f : __expf(mrow[r] - mnew);
        p[r] = (x[r] == -INFINITY) ? 0.0f : __expf(x[r] - mnew);
        mrow[r] = mnew;
      }
      float ps[8];
#pragma unroll
      for (int r = 0; r < 8; r++) ps[r] = p[r];
#pragma unroll
      for (int off = 1; off <= 8; off <<= 1)
#pragma unroll
        for (int r = 0; r < 8; r++) ps[r] += __shfl_xor(ps[r], off, 32);
#pragma unroll
      for (int r = 0; r < 8; r++) srow[r] = srow[r] * alpha[r] + ps[r];
#pragma unroll
      for (int g = 0; g < 8; g++)
#pragma unroll
        for (int r = 0; r < 8; r++) O[g][r] *= alpha[r];

      // stage P through LDS to re-fragment C-layout -> A-layout
#pragma unroll
      for (int r = 0; r < 8; r++) Plds[w][(r + 8 * hi) * 16 + lo] = p[r];
      __syncthreads();

      // O += P * Vpool : 4 K-chunks x 8 N-groups of fp32 WMMAs
#pragma unroll
      for (int kk = 0; kk < 4; kk++) {
        v2f ap;                         // A 16x4: M = lo, K = 4kk+v+2*hi
        ap.x = Plds[w][lo * 16 + 4 * kk + 2 * hi];
        ap.y = Plds[w][lo * 16 + 4 * kk + 2 * hi + 1];
#pragma unroll
        for (int g = 0; g < 8; g++) {
          v2f bv;                       // B 4x16: K = 4kk+v+2*hi, N = g*16+lo
          bv.x = Vlds[(4 * kk + 2 * hi) * DIM + g * 16 + lo];
          bv.y = Vlds[(4 * kk + 2 * hi + 1) * DIM + g * 16 + lo];
          O[g] = wmma4(ap, bv, O[g]);
        }
      }
    }
  }

  // normalize + write out
#pragma unroll
  for (int r = 0; r < 8; r++) {
    float inv = (srow[r] > 0.0f) ? 1.0f / srow[r] : 0.0f;
    size_t qglob = (size_t)(h * SEQ + ib * BLK + w * 16 + r + 8 * hi);
#pragma unroll
    for (int g = 0; g < 8; g++)
      out[qglob * DIM + g * 16 + lo] = O[g][r] * inv;
  }
}

// ---------------------------------------------------------------------------
extern "C" void kernel_launch(void* const* d_in, const int* in_sizes, int n_in,
                              void* d_out, int out_size, void* d_ws, size_t ws_size,
                              hipStream_t stream) {
  const float* q = (const float*)d_in[0];
  const float* k = (const float*)d_in[1];
  const float* v = (const float*)d_in[2];
  float* out = (float*)d_out;

  // workspace layout (floats): qg | kg | probs | est | sim(int) | msk(int)
  float* qg = (float*)d_ws;                       // 8*128*128
  float* kg = qg + NH * NCH * DIM;                // 8*128*128
  float* probs = kg + NH * NCH * DIM;             // 8*128*128
  float* est = probs + NH * NCH * NCH;            // 8*16*16
  int* sim = (int*)(est + NH * SBK * SBK);        // 8*16
  int* msk = sim + NH * SBK;                      // 8*16*16

  hipLaunchKernelGGL(pool16_kernel, dim3(NH * NCH), dim3(128), 0, stream, q, k, qg, kg);
  hipLaunchKernelGGL(chunk_softmax_kernel, dim3(NH * NCH), dim3(128), 0, stream, qg, kg, probs);
  hipLaunchKernelGGL(est_pool_kernel, dim3(NH), dim3(256), 0, stream, probs, est);
  hipLaunchKernelGGL(ksim_kernel, dim3(NH * SBK), dim3(64), 0, stream, k, sim);
  hipLaunchKernelGGL(mask_kernel, dim3(1), dim3(128), 0, stream, est, sim, msk);
  hipLaunchKernelGGL(attn_kernel, dim3(NH * SBK), dim3(256), 0, stream, q, k, v, msk, out);
}